// InstanceRigidModelClassic_40243843563529
// MI455X (gfx1250) — compile-verified
//
#include <hip/hip_runtime.h>

// ---------------------------------------------------------------------------
// Rigid 3D warp + trilinear resample, 256^3 fp32.
// Memory-bound: ~134 MB HBM traffic -> ~5.8us floor @ 23.3 TB/s (volume fits
// in the 192 MB L2, so gathers are L2 hits; HBM sees one read + one write).
// Coordinate transform runs on the matrix pipe: V_WMMA_F32_16X16X4_F32.
// All gather/store indices are 32-bit (256^3 = 2^24 elements) so addressing
// uses SGPR-base + 32-bit VGPR-offset (GVS) instead of 64-bit VALU pairs.
// ---------------------------------------------------------------------------

typedef __attribute__((ext_vector_type(2))) float v2f;
typedef __attribute__((ext_vector_type(8))) float v8f;

#define DIMS 256

// ---------------- tiny 1-thread kernel: build the 4x4 transform ------------

__device__ __forceinline__ void mm3(const float* A, const float* B, float* C) {
    for (int r = 0; r < 3; ++r)
        for (int c = 0; c < 3; ++c)
            C[r * 3 + c] = A[r * 3 + 0] * B[0 * 3 + c] +
                           A[r * 3 + 1] * B[1 * 3 + c] +
                           A[r * 3 + 2] * B[2 * 3 + c];
}

__device__ __forceinline__ void mm4(const float* A, const float* B, float* C) {
    for (int r = 0; r < 4; ++r)
        for (int c = 0; c < 4; ++c)
            C[r * 4 + c] = A[r * 4 + 0] * B[0 * 4 + c] +
                           A[r * 4 + 1] * B[1 * 4 + c] +
                           A[r * 4 + 2] * B[2 * 4 + c] +
                           A[r * 4 + 3] * B[3 * 4 + c];
}

__global__ void build_T_kernel(const float* __restrict__ angle,
                               const float* __restrict__ trans,
                               const float* __restrict__ ref_v2r,
                               const float* __restrict__ flo_v2r,
                               const float* __restrict__ cog,
                               float* __restrict__ T_out) {
    if (threadIdx.x != 0 || blockIdx.x != 0) return;

    const float ax = angle[0], ay = angle[1], az = angle[2];
    const float cx = cosf(ax), sx = sinf(ax);
    const float cy = cosf(ay), sy = sinf(ay);
    const float cz = cosf(az), sz = sinf(az);

    // R = Rx @ Ry @ Rz (matches _compute_rotation)
    const float Rx[9] = {1, 0, 0, 0, cx, -sx, 0, sx, cx};
    const float Ry[9] = {cy, 0, sy, 0, 1, 0, -sy, 0, cy};
    const float Rz[9] = {cz, -sz, 0, sz, cz, 0, 0, 0, 1};
    float M1[9], R[9];
    mm3(Rx, Ry, M1);
    mm3(M1, Rz, R);

    // T_rig = Tci @ Tt @ Tr @ Tc = [ R | t + cog - R@cog ; 0 0 0 1 ]
    float Trig[16] = {0};
    for (int r = 0; r < 3; ++r) {
        Trig[r * 4 + 0] = R[r * 3 + 0];
        Trig[r * 4 + 1] = R[r * 3 + 1];
        Trig[r * 4 + 2] = R[r * 3 + 2];
        Trig[r * 4 + 3] = trans[r] + cog[r] -
                          (R[r * 3 + 0] * cog[0] + R[r * 3 + 1] * cog[1] + R[r * 3 + 2] * cog[2]);
    }
    Trig[15] = 1.0f;

    // inv(flo_v2r) — affine inverse (last row [0,0,0,1])
    const float a00 = flo_v2r[0], a01 = flo_v2r[1], a02 = flo_v2r[2],  b0 = flo_v2r[3];
    const float a10 = flo_v2r[4], a11 = flo_v2r[5], a12 = flo_v2r[6],  b1 = flo_v2r[7];
    const float a20 = flo_v2r[8], a21 = flo_v2r[9], a22 = flo_v2r[10], b2 = flo_v2r[11];
    const float det = a00 * (a11 * a22 - a12 * a21)
                    - a01 * (a10 * a22 - a12 * a20)
                    + a02 * (a10 * a21 - a11 * a20);
    const float id = 1.0f / det;
    float Fi[16] = {0};
    Fi[0]  = (a11 * a22 - a12 * a21) * id;
    Fi[1]  = (a02 * a21 - a01 * a22) * id;
    Fi[2]  = (a01 * a12 - a02 * a11) * id;
    Fi[4]  = (a12 * a20 - a10 * a22) * id;
    Fi[5]  = (a00 * a22 - a02 * a20) * id;
    Fi[6]  = (a02 * a10 - a00 * a12) * id;
    Fi[8]  = (a10 * a21 - a11 * a20) * id;
    Fi[9]  = (a01 * a20 - a00 * a21) * id;
    Fi[10] = (a00 * a11 - a01 * a10) * id;
    Fi[3]  = -(Fi[0] * b0 + Fi[1] * b1 + Fi[2] * b2);
    Fi[7]  = -(Fi[4] * b0 + Fi[5] * b1 + Fi[6] * b2);
    Fi[11] = -(Fi[8] * b0 + Fi[9] * b1 + Fi[10] * b2);
    Fi[15] = 1.0f;

    // T = inv(flo) @ T_rig @ ref
    float Tmp[16], T[16];
    mm4(Trig, ref_v2r, Tmp);
    mm4(Fi, Tmp, T);
    for (int q = 0; q < 16; ++q) T_out[q] = T[q];
}

// ---------------- main kernel: WMMA coord transform + trilinear gather -----
//
// Block = 256 threads (8 waves) = one (i,j) output line along k.
// Each wave owns 32 consecutive k. Per wave:
//   A(16x4)  : row M = T[M%4, :]   (ISA layout: lanes 0-15 hold K=0,1;
//                                   lanes 16-31 hold K=2,3)
//   B(4x16)  : column N = [i, j, k0+N, 1]^T  (lanes 0-15 hold rows K=0,1;
//                                             lanes 16-31 hold rows K=2,3)
//   D(16x16) : row 0 = di, row 1 = dj, row 2 = dk for 16 voxels.
// Two WMMAs cover k0..k0+31; lane L<16 reads D1 rows 0-2 (M=0..2, N=L),
// lane L>=16 reads D2 rows 8-10 (M=8..10 == rows 0-2 replicated, N=L-16),
// so every lane ends with the coords of voxel k0+L.

__global__ void __launch_bounds__(256)
warp_trilinear_kernel(const float* __restrict__ vol,
                      const float* __restrict__ T,
                      float* __restrict__ out) {
    const int k = threadIdx.x;

    const int lane = threadIdx.x & 31;
    const int k0   = k & ~31;          // wave's first voxel along k
    const int half = lane >> 4;        // 0: lanes 0-15, 1: lanes 16-31
    const int r    = lane & 3;         // T row replicated mod 4

    // A fragment: 2 VGPRs per lane (K pair selected by lane half)
    v2f a;
    a.x = T[r * 4 + 2 * half];
    a.y = T[r * 4 + 2 * half + 1];

    // B fragments (branch-free so EXEC stays all-ones into the WMMAs)
    const float fi_c = (float)(blockIdx.x >> 8);        // i
    const float fj_c = (float)(blockIdx.x & (DIMS - 1)); // j
    v2f b1, b2;
    b1.x = half ? (float)(k0 + lane - 16) : fi_c;   // K=0 row: i   | K=2 row: k
    b1.y = half ? 1.0f                    : fj_c;   // K=1 row: j   | K=3 row: 1
    b2.x = half ? (float)(k0 + lane)      : fi_c;
    b2.y = half ? 1.0f                    : fj_c;

    v8f c = {0.f, 0.f, 0.f, 0.f, 0.f, 0.f, 0.f, 0.f};
    v8f d1 = __builtin_amdgcn_wmma_f32_16x16x4_f32(false, a, false, b1, (short)0, c, false, false);
    v8f d2 = __builtin_amdgcn_wmma_f32_16x16x4_f32(false, a, false, b2, (short)0, c, false, false);

    float di = half ? d2[0] : d1[0];
    float dj = half ? d2[1] : d1[1];
    float dk = half ? d2[2] : d1[2];

    // ---- trilinear interpolation, zero outside bounds ----
    const float fmax_ = (float)(DIMS - 1);
    const bool ok = (di >= 0.f) && (di <= fmax_) &&
                    (dj >= 0.f) && (dj <= fmax_) &&
                    (dk >= 0.f) && (dk <= fmax_);

    di = fminf(fmaxf(di, 0.f), fmax_);
    dj = fminf(fmaxf(dj, 0.f), fmax_);
    dk = fminf(fmaxf(dk, 0.f), fmax_);

    const float ffi = floorf(di), ffj = floorf(dj), ffk = floorf(dk);
    const unsigned i0 = (unsigned)(int)ffi;
    const unsigned j0 = (unsigned)(int)ffj;
    const unsigned q0 = (unsigned)(int)ffk;
    const unsigned i1 = min(i0 + 1u, (unsigned)(DIMS - 1));
    const unsigned j1 = min(j0 + 1u, (unsigned)(DIMS - 1));
    const unsigned q1 = min(q0 + 1u, (unsigned)(DIMS - 1));
    const float wi = di - ffi, wj = dj - ffj, wk = dk - ffk;

    // 32-bit flat element indices: (i*256 + j)*256 + k == (i<<16)|(j<<8)|k.
    // Indexing with unsigned keeps addressing in SGPR-base + 32-bit voffset.
    const unsigned r00 = (i0 << 16) | (j0 << 8);
    const unsigned r01 = (i0 << 16) | (j1 << 8);
    const unsigned r10 = (i1 << 16) | (j0 << 8);
    const unsigned r11 = (i1 << 16) | (j1 << 8);

    // Pull the farthest line toward the caches early (global_prefetch_b8)
    __builtin_prefetch(vol + (r11 + q0), 0, 0);

    const float v000 = vol[r00 + q0], v001 = vol[r00 + q1];
    const float v010 = vol[r01 + q0], v011 = vol[r01 + q1];
    const float v100 = vol[r10 + q0], v101 = vol[r10 + q1];
    const float v110 = vol[r11 + q0], v111 = vol[r11 + q1];

    const float c00 = v000 + wk * (v001 - v000);
    const float c01 = v010 + wk * (v011 - v010);
    const float c10 = v100 + wk * (v101 - v100);
    const float c11 = v110 + wk * (v111 - v110);
    const float c0  = c00 + wj * (c01 - c00);
    const float c1  = c10 + wj * (c11 - c10);
    const float val = c0 + wi * (c1 - c0);

    // Coalesced store: block-uniform base + lane offset, 32-bit index.
    out[(blockIdx.x << 8) | (unsigned)k] = ok ? val : 0.0f;
}

// ---------------------------------------------------------------------------

extern "C" void kernel_launch(void* const* d_in, const int* in_sizes, int n_in,
                              void* d_out, int out_size, void* d_ws, size_t ws_size,
                              hipStream_t stream) {
    const float* vol   = (const float*)d_in[0];  // image_targ [1,1,256,256,256]
    const float* angle = (const float*)d_in[1];  // [1,3]
    const float* trans = (const float*)d_in[2];  // [1,3]
    const float* refm  = (const float*)d_in[3];  // [4,4]
    const float* flom  = (const float*)d_in[4];  // [4,4]
    const float* cog   = (const float*)d_in[5];  // [3]

    float* T = (float*)d_ws;                     // 16 floats of scratch

    build_T_kernel<<<1, 32, 0, stream>>>(angle, trans, refm, flom, cog, T);
    warp_trilinear_kernel<<<DIMS * DIMS, DIMS, 0, stream>>>(vol, T, (float*)d_out);
}